// Embed_88725434401528
// MI455X (gfx1250) — compile-verified
//
#include <hip/hip_runtime.h>

// Problem constants from the reference
#define BDIM   16
#define LDIM   100
#define LOCDIM 2000
#define EDIM   16
#define NTILE  (LOCDIM / 16)   // 125 tiles of 16 locations
#define NWAVE  8               // 256 threads / wave32

typedef __attribute__((ext_vector_type(2))) float v2f;
typedef __attribute__((ext_vector_type(8))) float v8f;

// One block per (b, l). Each wave processes 16x16 (loc x e) output tiles via
// V_WMMA_F32_16X16X4_F32 as a rank-1 update: D = s(16x1) * Acoef(1x16) + C,
// where C broadcasts Bconst[e] over the 16 rows. Store-bandwidth bound
// (~205 MB of NT stores, mat2 reads stay L2-resident).
__global__ __launch_bounds__(256) void stlstm_embed_kernel(
    const int*   __restrict__ traj_loc,   // (B, L)
    const float* __restrict__ mat2,       // (LOC, LOC)
    const float* __restrict__ vec,        // (B, L)
    const int*   __restrict__ traj_len,   // (B,)
    const float* __restrict__ emb_sl,     // (2, E)
    const float* __restrict__ emb_su,     // (2, E)
    const float* __restrict__ emb_tl,     // (2, E)
    const float* __restrict__ emb_tu,     // (2, E)
    float*       __restrict__ out)        // (B, L, LOC, E)
{
    const int blk  = blockIdx.x;            // b*L + l
    const int b    = blk / LDIM;
    const int l    = blk - b * LDIM;
    const int tid  = threadIdx.x;
    const int lane = tid & 31;
    const int wave = tid >> 5;
    const int e    = lane & 15;

    // Uniform per-(b,l) scalars
    const bool  valid = l < traj_len[b];
    const int   m     = valid ? 1 : 0;
    const float t     = vec[blk];
    int idx = traj_loc[blk] - 1;
    idx = idx < 0 ? 0 : (idx > (LOCDIM - 1) ? (LOCDIM - 1) : idx);

    // Per-e coefficients (lane e and lane e+16 compute the same values)
    const float esl = emb_sl[m * EDIM + e];
    const float esu = emb_su[m * EDIM + e];
    const float etl = emb_tl[m * EDIM + e];
    const float etu = emb_tu[m * EDIM + e];

    // out[loc,e] = Acoef[e] * s_loc + Bconst[e]
    const float Acoef  = (esu - esl) * (1.0f / 500.0f);
    const float Bconst = esl + (etl * (3600.0f - t) + etu * t) * (1.0f / 3600.0f);

    const float* __restrict__ row = mat2 + (long)idx * LOCDIM;
    float* __restrict__ outBase   = out + (long)blk * (LOCDIM * EDIM);

    // B operand (4x16 f32, 2 VGPRs). Only the K=0 row (VGPR0, lanes 0-15) is
    // meaningful; we fold BOTH the validity mask and the "K>=1 rows are zero"
    // requirement into it:
    //   - invalid (b,l): Acoef_eff = 0  ->  D = Bconst (delta_s treated as 0)
    //   - lanes 16-31 (K=1/2 row of B) = 0 -> A's K=2 entries may hold garbage
    // This keeps the inner loop completely branch/cndmask-free.
    v2f bm;
    bm.x = (valid && lane < 16) ? Acoef : 0.0f;
    bm.y = 0.0f;

    // C operand: Bconst[N] for every M row -> every accumulator VGPR = Bconst
    v8f c;
#pragma unroll
    for (int r = 0; r < 8; ++r) c[r] = Bconst;

    // Per-lane store base: lane n writes column N=n&15 of rows M=r+8*(n>=16)
    float* __restrict__ pBaseLane = outBase + (long)((lane >> 4) * 8) * EDIM + e;

    for (int tIdx = wave; tIdx < NTILE; tIdx += NWAVE) {
        const int loc0 = tIdx * 16;

        // Speculative prefetch of the next tile's slice (no guard: speculative
        // prefetch past the row tail is dropped silently, and this avoids any
        // EXEC manipulation next to the WMMA, which requires EXEC == all-1s).
        __builtin_prefetch(row + loc0 + NWAVE * 16 + e, 0, 0);

        // A operand (16x4 f32, 2 VGPRs): VGPR0 lanes 0-15 = s[M]; lanes 16-31
        // are K=2 entries that multiply B's zeroed rows, so the raw (possibly
        // duplicated) load result is fine unmasked. Load is always in-bounds
        // (idx clamped); invalidity is already folded into bm.
        v2f a;
        a.x = row[loc0 + e];
        a.y = 0.0f;

        // D = A x B + C  (16x16 f32 accumulators, 8 VGPRs)
        v8f d = __builtin_amdgcn_wmma_f32_16x16x4_f32(
            /*neg_a=*/false, a, /*neg_b=*/false, bm,
            /*c_mod=*/(short)0, c, /*reuse_a=*/false, /*reuse_b=*/false);

        // Store: VGPR r, lane n -> out[loc0 + r + 8*(n>=16)][n&15].
        // Non-temporal: output is streamed once and never re-read; keep the
        // 192MB L2 for mat2 reuse instead of rinsing it with 205MB of stores.
        float* __restrict__ p = pBaseLane + (long)loc0 * EDIM;
#pragma unroll
        for (int r = 0; r < 8; ++r)
            __builtin_nontemporal_store(d[r], p + (long)r * EDIM);
    }
}

extern "C" void kernel_launch(void* const* d_in, const int* in_sizes, int n_in,
                              void* d_out, int out_size, void* d_ws, size_t ws_size,
                              hipStream_t stream) {
    (void)in_sizes; (void)n_in; (void)out_size; (void)d_ws; (void)ws_size;
    const int*   traj_loc = (const int*)  d_in[0];
    const float* mat2     = (const float*)d_in[1];
    const float* vec      = (const float*)d_in[2];
    const int*   traj_len = (const int*)  d_in[3];
    const float* emb_sl   = (const float*)d_in[4];
    const float* emb_su   = (const float*)d_in[5];
    const float* emb_tl   = (const float*)d_in[6];
    const float* emb_tu   = (const float*)d_in[7];
    float* out = (float*)d_out;

    dim3 grid(BDIM * LDIM);   // 1600 blocks, one per (b, l)
    dim3 block(256);          // 8 wave32s
    stlstm_embed_kernel<<<grid, block, 0, stream>>>(
        traj_loc, mat2, vec, traj_len, emb_sl, emb_su, emb_tl, emb_tu, out);
}